// LightGCNModel_36395552866748
// MI455X (gfx1250) — compile-verified
//
#include <hip/hip_runtime.h>

typedef float v2f __attribute__((ext_vector_type(2)));
typedef float v8f __attribute__((ext_vector_type(8)));

#define EMB_D 64

// ---------------------------------------------------------------------------
// init: x = acc = concat(user_emb, item_emb), float4 grid-stride (HBM stream)
// ---------------------------------------------------------------------------
__global__ void lgcn_init(const float* __restrict__ ue, const float* __restrict__ ie,
                          float* __restrict__ x, float* __restrict__ acc,
                          long n_user_f, long total_f) {
    long tid    = (long)blockIdx.x * blockDim.x + threadIdx.x;
    long stride = (long)gridDim.x * blockDim.x * 4;
    for (long j = tid * 4; j < total_f; j += stride) {
        float4 v = (j < n_user_f) ? *(const float4*)(ue + j)
                                  : *(const float4*)(ie + (j - n_user_f));
        *(float4*)(x + j)   = v;
        *(float4*)(acc + j) = v;
    }
}

__global__ void lgcn_zero(float* __restrict__ y, long total_f) {
    long tid    = (long)blockIdx.x * blockDim.x + threadIdx.x;
    long stride = (long)gridDim.x * blockDim.x * 4;
    float4 z = {0.f, 0.f, 0.f, 0.f};
    for (long j = tid * 4; j < total_f; j += stride)
        *(float4*)(y + j) = z;
}

// ---------------------------------------------------------------------------
// SpMM scatter: one wave32 per edge. Lane k handles floats [2k,2k+1] of the
// 64-float row: coalesced 256B float2 gather of x[src] (L2-resident, 38.4MB),
// then 2 lane-parallel global_atomic_add_f32 into y[dst].
// ---------------------------------------------------------------------------
__global__ void lgcn_scatter(const float* __restrict__ x, float* __restrict__ y,
                             const int* __restrict__ src, const int* __restrict__ dst,
                             const float* __restrict__ w, int n_edges) {
    int gtid = blockIdx.x * blockDim.x + threadIdx.x;
    int e    = gtid >> 5;          // edge index (wave-uniform)
    int lane = threadIdx.x & 31;
    if (e >= n_edges) return;

    int   s  = src[e];
    int   t  = dst[e];
    float ww = w[e];

    // warm L2 for a source row ~2048 edges ahead (gfx1250 global_prefetch path)
    int e2 = e + 2048;
    if (e2 < n_edges) {
        int s2 = src[e2];
        __builtin_prefetch(x + (size_t)s2 * EMB_D + lane * 2, 0, 1);
    }

    const float* xs = x + (size_t)s * EMB_D + lane * 2;
    float*       yd = y + (size_t)t * EMB_D + lane * 2;
    v2f xv = *(const v2f*)xs;
    atomicAdd(yd + 0, ww * xv.x);
    atomicAdd(yd + 1, ww * xv.y);
}

__global__ void lgcn_accadd(float* __restrict__ acc, const float* __restrict__ y,
                            long total_f) {
    long tid    = (long)blockIdx.x * blockDim.x + threadIdx.x;
    long stride = (long)gridDim.x * blockDim.x * 4;
    for (long j = tid * 4; j < total_f; j += stride) {
        float4 a = *(const float4*)(acc + j);
        float4 b = *(const float4*)(y + j);
        a.x += b.x; a.y += b.y; a.z += b.z; a.w += b.w;
        *(float4*)(acc + j) = a;
    }
}

// ---------------------------------------------------------------------------
// Scoring via V_WMMA_F32_16X16X4_F32: one wave per 16 pairs.
// A = 16 user rows (16xK), B = (16 item rows)^T (Kx16); accumulate K=64 in
// 16 chained WMMAs, then take the diagonal of the 16x16 f32 result.
// Per ISA VGPR layouts, lane l supplies A(M=l&15, K=2*(l>>4)+{0,1}) and
// B(K=2*(l>>4)+{0,1}, N=l&15) -> identical per-lane row/K addressing.
// out = (acc_u . acc_v) / 16  (the two /(L+1)=/4 factors).
// ---------------------------------------------------------------------------
__global__ void lgcn_score_wmma(const float* __restrict__ acc,
                                const int* __restrict__ users,
                                const int* __restrict__ items,
                                int n_users, float* __restrict__ out, int batch) {
    const int lane = threadIdx.x;      // 0..31, blockDim.x == 32
    const int m    = lane & 15;        // pair-in-tile for A-row / B-col
    const int hf   = lane >> 4;        // lane half selects K pair {0,1} vs {2,3}
    const int base = blockIdx.x * 16;

    int p  = base + m;
    int pc = (p < batch) ? p : (batch - 1);   // clamp: keep EXEC all-1s for WMMA

    const float* urow = acc + (size_t)users[pc] * EMB_D;
    const float* vrow = acc + ((size_t)n_users + (size_t)items[pc]) * EMB_D;

    v8f c = {0.f, 0.f, 0.f, 0.f, 0.f, 0.f, 0.f, 0.f};
    const int ko = 2 * hf;
#pragma unroll
    for (int kb = 0; kb < EMB_D; kb += 4) {
        v2f a = *(const v2f*)(urow + kb + ko);
        v2f b = *(const v2f*)(vrow + kb + ko);
        c = __builtin_amdgcn_wmma_f32_16x16x4_f32(
                /*neg_a=*/false, a, /*neg_b=*/false, b,
                /*c_mod=*/(short)0, c, /*reuse_a=*/false, /*reuse_b=*/false);
    }

    // Diagonal extraction: (M=N=m) lives in lane m (VGPR m) for m<8,
    // and lane 16+m (VGPR m-8) for m>=8.
    bool has = (lane < 8) || (lane >= 24);
    int  idx = (lane < 8) ? lane : (lane - 24);
    float val = (idx == 0) ? c[0] :
                (idx == 1) ? c[1] :
                (idx == 2) ? c[2] :
                (idx == 3) ? c[3] :
                (idx == 4) ? c[4] :
                (idx == 5) ? c[5] :
                (idx == 6) ? c[6] : c[7];
    if (has && p < batch)
        out[p] = val * 0.0625f;   // (1/4)*(1/4)
}

// ---------------------------------------------------------------------------
// Host launcher
// ---------------------------------------------------------------------------
extern "C" void kernel_launch(void* const* d_in, const int* in_sizes, int n_in,
                              void* d_out, int out_size, void* d_ws, size_t ws_size,
                              hipStream_t stream) {
    const float* user_emb = (const float*)d_in[0];
    const float* item_emb = (const float*)d_in[1];
    const int*   edge_src = (const int*)  d_in[2];
    const int*   edge_dst = (const int*)  d_in[3];
    const float* edge_w   = (const float*)d_in[4];
    const int*   users    = (const int*)  d_in[5];
    const int*   items    = (const int*)  d_in[6];
    float*       out      = (float*)d_out;

    const int  n_users  = in_sizes[0] / EMB_D;
    const int  n_items  = in_sizes[1] / EMB_D;
    const int  n_edges  = in_sizes[2];
    const int  batch    = in_sizes[5];
    const long n_user_f = (long)n_users * EMB_D;
    const long total_f  = (long)(n_users + n_items) * EMB_D;

    // workspace: x_a | x_b | acc  (each total_f floats; total_f*4 is 256B-mult)
    float* x_a   = (float*)d_ws;
    float* x_b   = x_a + total_f;
    float* accum = x_b + total_f;

    const int TB = 256;
    const int streamBlocks = 2048;

    lgcn_init<<<streamBlocks, TB, 0, stream>>>(user_emb, item_emb, x_a, accum,
                                               n_user_f, total_f);

    float* cur = x_a;
    float* nxt = x_b;
    for (int layer = 0; layer < 3; ++layer) {
        lgcn_zero<<<streamBlocks, TB, 0, stream>>>(nxt, total_f);
        long scatterThreads = (long)n_edges * 32;
        int  scatterBlocks  = (int)((scatterThreads + TB - 1) / TB);
        lgcn_scatter<<<scatterBlocks, TB, 0, stream>>>(cur, nxt, edge_src,
                                                       edge_dst, edge_w, n_edges);
        lgcn_accadd<<<streamBlocks, TB, 0, stream>>>(accum, nxt, total_f);
        float* tmp = cur; cur = nxt; nxt = tmp;
    }

    int scoreBlocks = (batch + 15) / 16;
    lgcn_score_wmma<<<scoreBlocks, 32, 0, stream>>>(accum, users, items,
                                                    n_users, out, batch);
}